// CudaTensorProduct_18674517803223
// MI455X (gfx1250) — compile-verified
//
#include <hip/hip_runtime.h>
#include <stdint.h>

// ---------------------------------------------------------------------------
// e3nn CG tensor product on MI455X, v4: scalar-pipe tables + async staging,
// NT output stores. (Main kernel placed first so disasm shows the hot loop.)
//   out[b, j] = sum_{k in seg[j]..seg[j+1]} in1[b,idx1[k]] * in2[b,idx2[k]] * cb[k]
// B=16384, d1=d2=32, D=1024. Memory floor ~3us (64MB out @ 23.3 TB/s).
//
// Mapping: lane = batch row (2 rows/lane, 64 rows/block); j wave-uniform.
//  - seg bounds + packed entries via s_load (readfirstlane-forced uniformity)
//    -> unpack + loop control on SALU pipe, co-issued with VALU/LDS.
//  - gathers: conflict-free LDS (33-dword row stride = lane-bank permutation),
//    +32-row operand folded into the DS 16-bit immediate (+4224B).
//  - output staged in 64x128 LDS tile, flushed as NT b128 (512B/wave, no L2
//    pollution of the shared tables).
//  - input rows staged with global_load_async_to_lds_b32 + s_wait_asynccnt.
// WMMA intentionally unused: dense recast = 34 GFLOP fp32 vs 0.4 GFLOP real,
// ~50x slower than the bandwidth floor.
// ---------------------------------------------------------------------------

#define THREADS  256               // 8 waves
#define BT       64                // batch rows per block (2 per lane)
#define D_OUT    1024
#define JSPLIT   2                 // column-space split (grid.y)
#define JBLK     (D_OUT / JSPLIT)  // 512 columns per block
#define JC       128               // columns per LDS tile chunk
#define ROWP     33                // padded input row stride (dwords)
#define TILEP    132               // padded tile row stride (dwords, 16B-aligned)

typedef float v4f __attribute__((ext_vector_type(4)));

__device__ __forceinline__ uint32_t lds_addr_of(const void* p) {
  return (uint32_t)(uintptr_t)p;  // low 32 bits of generic LDS pointer == LDS offset
}

__device__ __forceinline__ void async_ld_b32(uint32_t lds, const void* gptr) {
  asm volatile("global_load_async_to_lds_b32 %0, %1, off"
               :: "v"(lds), "v"((uint64_t)(uintptr_t)gptr) : "memory");
}

__device__ __forceinline__ void wait_async_zero() {
  asm volatile("s_wait_asynccnt 0" ::: "memory");
}

// ---------------------------------------------------------------------------
// Main kernel (first in module so the disasm snippet shows the hot loop).
// ---------------------------------------------------------------------------
__global__ __launch_bounds__(THREADS) void tp_main_kernel(
    const float* __restrict__ in1, const float* __restrict__ in2,
    const int* __restrict__ seg_g, const uint2* __restrict__ ent_g,
    float* __restrict__ out, int B) {

  __shared__ __align__(16) float s_in1[BT * ROWP];    //  8.4 KB
  __shared__ __align__(16) float s_in2[BT * ROWP];    //  8.4 KB
  __shared__ __align__(16) float s_tile[BT * TILEP];  // 33.8 KB

  const int tid   = threadIdx.x;
  const int lane  = tid & 31;
  const int wv    = tid >> 5;
  const int b0    = blockIdx.x * BT;
  const int jbase = blockIdx.y * JBLK;

  // ---- async stage BT input rows into padded LDS (ASYNCcnt tracked) ----
  for (int i = tid; i < BT * 32; i += THREADS) {
    const int row = i >> 5, m = i & 31;
    if (b0 + row < B) {
      async_ld_b32(lds_addr_of(&s_in1[row * ROWP + m]),
                   (const void*)(in1 + (size_t)(b0 + row) * 32 + m));
      async_ld_b32(lds_addr_of(&s_in2[row * ROWP + m]),
                   (const void*)(in2 + (size_t)(b0 + row) * 32 + m));
    }
  }
  wait_async_zero();
  __syncthreads();

  const char* abase = (const char*)(s_in1 + lane * ROWP);
  const char* bbase = (const char*)(s_in2 + lane * ROWP);

  for (int c = 0; c < JBLK / JC; ++c) {          // 4 chunks of 128 columns
    // ---- compute: wave-uniform column j; lanes = batch rows ----
    for (int jj = 0; jj < JC / 8; ++jj) {        // 16 columns per wave
      const int jcol = wv * (JC / 8) + jj;       // 0..127 in chunk
      const int ju   = __builtin_amdgcn_readfirstlane(jbase + c * JC + jcol);
      const int s = __builtin_amdgcn_readfirstlane(seg_g[ju]);     // s_load
      const int e = __builtin_amdgcn_readfirstlane(seg_g[ju + 1]); // s_load
      float acc0 = 0.f, acc1 = 0.f;
      int k = s;
      for (; k + 2 <= e; k += 2) {               // 2x unroll: amortize kmcnt
        const uint2 e0 = ent_g[k];               // s_load_b64 (uniform)
        const uint2 e1 = ent_g[k + 1];
        {
          const uint32_t o1 = e0.x & 0xffffu, o2 = e0.x >> 16;  // SALU
          const float cbv = __uint_as_float(e0.y);
          const float* pa = (const float*)(abase + o1);
          const float* pb = (const float*)(bbase + o2);
          acc0 = fmaf(pa[0] * pb[0], cbv, acc0);
          acc1 = fmaf(pa[32 * ROWP] * pb[32 * ROWP], cbv, acc1);  // +4224B imm
        }
        {
          const uint32_t o1 = e1.x & 0xffffu, o2 = e1.x >> 16;
          const float cbv = __uint_as_float(e1.y);
          const float* pa = (const float*)(abase + o1);
          const float* pb = (const float*)(bbase + o2);
          acc0 = fmaf(pa[0] * pb[0], cbv, acc0);
          acc1 = fmaf(pa[32 * ROWP] * pb[32 * ROWP], cbv, acc1);
        }
      }
      if (k < e) {
        const uint2 e0 = ent_g[k];
        const uint32_t o1 = e0.x & 0xffffu, o2 = e0.x >> 16;
        const float cbv = __uint_as_float(e0.y);
        const float* pa = (const float*)(abase + o1);
        const float* pb = (const float*)(bbase + o2);
        acc0 = fmaf(pa[0] * pb[0], cbv, acc0);
        acc1 = fmaf(pa[32 * ROWP] * pb[32 * ROWP], cbv, acc1);
      }
      s_tile[lane * TILEP + jcol]        = acc0;
      s_tile[(lane + 32) * TILEP + jcol] = acc1;
    }
    __syncthreads();

    // ---- flush tile: NT float4 -> b128, 512B contiguous per wave ----
#pragma unroll
    for (int q = 0; q < 8; ++q) {
      const int idx = tid + THREADS * q;  // 0..2047
      const int r   = idx >> 5;           // 0..63
      const int c4  = idx & 31;           // 0..31 float4 slots per row
      if (b0 + r < B) {
        const v4f v = *(const v4f*)&s_tile[r * TILEP + c4 * 4];
        v4f* dst = (v4f*)(out + (size_t)(b0 + r) * D_OUT + jbase + c * JC + c4 * 4);
        __builtin_nontemporal_store(v, dst);  // write-once stream: NT hint
      }
    }
    __syncthreads();  // tile reusable next chunk
  }
}

// ---------------------------------------------------------------------------
// Prep: seg_start[j] = lower_bound(out_idx, j); pack entries {byte-offsets, cb}.
// pk = (idx1*4) | ((idx2*4) << 16) -> direct byte offsets into padded rows.
// ---------------------------------------------------------------------------
__global__ void tp_prep_kernel(const int* __restrict__ out_idx,
                               const int* __restrict__ idx1,
                               const int* __restrict__ idx2,
                               const float* __restrict__ cb,
                               int nnz, int D,
                               int* __restrict__ seg,
                               uint2* __restrict__ ent) {
  int i = blockIdx.x * blockDim.x + threadIdx.x;
  if (i <= D) {
    int lo = 0, hi = nnz;
    while (lo < hi) {
      int mid = (lo + hi) >> 1;
      if (out_idx[mid] < i) lo = mid + 1; else hi = mid;
    }
    seg[i] = lo;
  }
  if (i < nnz) {
    uint32_t pk = (uint32_t)((idx1[i] & 31) << 2) |
                  ((uint32_t)((idx2[i] & 31) << 2) << 16);
    ent[i] = make_uint2(pk, __float_as_uint(cb[i]));
  }
}

// ---------------------------------------------------------------------------
// Launch. Inputs: in1, in2, cb_vals, idx1, idx2, out_idx (setup_inputs order).
// ---------------------------------------------------------------------------
extern "C" void kernel_launch(void* const* d_in, const int* in_sizes, int n_in,
                              void* d_out, int out_size, void* d_ws, size_t ws_size,
                              hipStream_t stream) {
  const float* in1  = (const float*)d_in[0];
  const float* in2  = (const float*)d_in[1];
  const float* cb   = (const float*)d_in[2];
  const int*   idx1 = (const int*)d_in[3];
  const int*   idx2 = (const int*)d_in[4];
  const int*   oidx = (const int*)d_in[5];

  const int nnz = in_sizes[2];
  const int B   = in_sizes[0] / 32;
  const int D   = out_size / B;  // 1024

  int*   seg = (int*)d_ws;
  uint2* ent = (uint2*)((char*)d_ws + 8192);

  const int N = (nnz > D + 1) ? nnz : (D + 1);
  tp_prep_kernel<<<(N + 255) / 256, 256, 0, stream>>>(oidx, idx1, idx2, cb, nnz, D, seg, ent);

  dim3 grid((B + BT - 1) / BT, JSPLIT);
  tp_main_kernel<<<grid, THREADS, 0, stream>>>(in1, in2, seg, ent, (float*)d_out, B);
}